// EdgeConvUnit_42348377538668
// MI455X (gfx1250) — compile-verified
//
#include <hip/hip_runtime.h>
#include <hip/hip_bf16.h>
#include <math.h>

// ---------------------------------------------------------------------------
// EdgeConv unit for MI455X (gfx1250, wave32, WMMA)
//   B=8, N=4096, C_IN=C_OUT=64, K=16, edge_in = 131
// Pipeline:
//   knn_kernel    : per-batch brute-force KNN (coords in LDS, register top-16)
//   w1pad_kernel  : zero-pad W1 (131x64) -> W1p (132x64) in workspace, so the
//                   WMMA K-loop has NO boundary conditionals (no exec churn)
//   edgemlp_kernel: per-point 16x132 @ 132x64 and 16x64 @ 64x64 GEMMs on
//                   v_wmma_f32_16x16x4_f32 (full fp32, matches reference),
//                   exact GELU, max-pool over the 16 neighbors.
// ---------------------------------------------------------------------------

#define NPTS   4096
#define NBATCH 8
#define KNN    16
#define CIN    64
#define EDGE_K 132   // 131 padded to multiple of 4
#define WAVES  4     // waves (points) per block in the MLP kernel

typedef __attribute__((ext_vector_type(2))) float v2f;
typedef __attribute__((ext_vector_type(8))) float v8f;

// ------------------------------- KNN ---------------------------------------
__global__ __launch_bounds__(256) void knn_kernel(
    const float* __restrict__ coords, int* __restrict__ nbr_idx)
{
    __shared__ float sx[NPTS], sy[NPTS], sz[NPTS];

    const int b = blockIdx.y;
    const float* cb = coords + (size_t)b * NPTS * 3;

    for (int i = threadIdx.x; i < NPTS; i += 256) {
        sx[i] = cb[i * 3 + 0];
        sy[i] = cb[i * 3 + 1];
        sz[i] = cb[i * 3 + 2];
    }
    __syncthreads();

    const int n = blockIdx.x * 256 + threadIdx.x;
    const float qx = sx[n], qy = sy[n], qz = sz[n];
    const float sqn = qx * qx + qy * qy + qz * qz;

    float bd[KNN];
    int   bi[KNN];
#pragma unroll
    for (int j = 0; j < KNN; ++j) { bd[j] = 3.0e38f; bi[j] = 0; }

    for (int m = 0; m < NPTS; ++m) {
        const float xm = sx[m], ym = sy[m], zm = sz[m];
        const float sqm = xm * xm + ym * ym + zm * zm;
        const float dot = qx * xm + qy * ym + qz * zm;
        float d2 = sqn + sqm - 2.0f * dot;          // reference formula
        d2 = fmaxf(d2, 0.0f);                       // matches max(d2, 0)
        if (m != n && d2 < bd[KNN - 1]) {           // diagonal excluded (+BIG)
            float dd = d2; int ii = m;
#pragma unroll
            for (int j = 0; j < KNN; ++j) {         // fully-unrolled insertion
                if (dd < bd[j]) {
                    float td = bd[j]; int ti = bi[j];
                    bd[j] = dd; bi[j] = ii;
                    dd = td; ii = ti;
                }
            }
        }
    }

    int* op = nbr_idx + ((size_t)b * NPTS + n) * KNN;
#pragma unroll
    for (int j = 0; j < KNN; ++j) op[j] = bi[j];
}

// --------------------- W1 zero-pad (131x64 -> 132x64) ------------------------
__global__ __launch_bounds__(256) void w1pad_kernel(
    const float* __restrict__ W1, float* __restrict__ W1p)
{
    const int i = blockIdx.x * 256 + threadIdx.x;
    if (i < EDGE_K * 64)
        W1p[i] = (i < 131 * 64) ? W1[i] : 0.0f;
}

// ---------------------------- Edge MLP (WMMA) -------------------------------
__device__ __forceinline__ float gelu_exact(float x) {
    return 0.5f * x * (1.0f + erff(x * 0.70710678118654752440f));
}

__global__ __launch_bounds__(32 * WAVES) void edgemlp_kernel(
    const float* __restrict__ feats, const float* __restrict__ coords,
    const int*   __restrict__ nbr_idx,
    const float* __restrict__ W1p, const float* __restrict__ b1,
    const float* __restrict__ W2,  const float* __restrict__ b2,
    float* __restrict__ out)
{
    // Per-wave LDS scratch: edge matrix (16 x 132) and hidden h (16 x 64).
    __shared__ float sEdge[WAVES][16 * EDGE_K];
    __shared__ float sH[WAVES][16 * CIN];

    const int lane = threadIdx.x & 31;
    const int w    = threadIdx.x >> 5;
    const int p    = blockIdx.x * WAVES + w;      // global point id, 0..B*N-1
    const int b    = p >> 12;                     // / 4096
    const int n    = p & (NPTS - 1);

    float* E = sEdge[w];
    float* H = sH[w];

    const size_t bn = (size_t)b * NPTS + n;
    const float* fc = feats + bn * CIN;           // center features
    const float  cf0 = fc[lane];
    const float  cf1 = fc[lane + 32];

    int gl = 0;
    if (lane < KNN) gl = nbr_idx[(size_t)p * KNN + lane];

    float cc = 0.0f;
    if (lane < 3) cc = coords[bn * 3 + lane];     // center coord component

    // ---- Stage edge matrix: [ctr_f | nbr_f - ctr_f | nbr_c - ctr_c | 0pad]
    for (int j = 0; j < KNN; ++j) {
        const int g = __shfl(gl, j);
        const float* fg = feats + ((size_t)b * NPTS + g) * CIN;
        const float f0 = fg[lane];
        const float f1 = fg[lane + 32];
        E[j * EDGE_K +       lane] = cf0;
        E[j * EDGE_K + 32  + lane] = cf1;
        E[j * EDGE_K + 64  + lane] = f0 - cf0;
        E[j * EDGE_K + 96  + lane] = f1 - cf1;
        if (lane < 4) {
            float e = 0.0f;
            if (lane < 3)
                e = coords[((size_t)b * NPTS + g) * 3 + lane] - cc;
            E[j * EDGE_K + 128 + lane] = e;       // lane 3 writes the 0 pad
        }
    }

    // WMMA f32 16x16x4 fragment coordinates for this lane:
    //   A (16x4):  M = lane&15, K pair = (lane>>4)*2 .. +1
    //   B (4x16):  N col = lane&15, K rows = (lane>>4)*2 .. +1
    const int mrow  = lane & 15;
    const int khalf = (lane >> 4) << 1;
    const int col   = lane & 15;

    // ---- Layer 1: C1 = E(16x132) @ W1p(132x64) -----------------------------
    // W1p is zero-padded, so the body is branch-free: 1 ds_load_b64 (A-frag)
    // + 8 global_load_b32 (B-frags, L1-resident) + 4 WMMA per K-step.
    v8f acc[4] = {};
#pragma unroll 4
    for (int k0 = 0; k0 < EDGE_K; k0 += 4) {
        const int kk = k0 + khalf;
        v2f a;
        a.x = E[mrow * EDGE_K + kk];
        a.y = E[mrow * EDGE_K + kk + 1];
#pragma unroll
        for (int t = 0; t < 4; ++t) {
            const int c2 = t * 16 + col;
            v2f bf;
            bf.x = W1p[kk * 64 + c2];
            bf.y = W1p[(kk + 1) * 64 + c2];
            acc[t] = __builtin_amdgcn_wmma_f32_16x16x4_f32(
                false, a, false, bf, (short)0, acc[t], false, false);
        }
    }

    // ---- bias + exact GELU -> H (LDS, needed for cross-lane A relayout) ----
#pragma unroll
    for (int t = 0; t < 4; ++t) {
        const float bb = b1[t * 16 + col];
#pragma unroll
        for (int r = 0; r < 8; ++r) {
            const int M = r + ((lane >> 4) << 3);
            const float x = acc[t][r] + bb;
            H[M * CIN + t * 16 + col] = gelu_exact(x);
        }
    }

    // ---- Layer 2: C2 = H(16x64) @ W2(64x64) --------------------------------
    v8f acc2[4] = {};
#pragma unroll 4
    for (int k0 = 0; k0 < CIN; k0 += 4) {
        const int kk = k0 + khalf;
        v2f a;
        a.x = H[mrow * CIN + kk];
        a.y = H[mrow * CIN + kk + 1];
#pragma unroll
        for (int t = 0; t < 4; ++t) {
            const int c2 = t * 16 + col;
            v2f bf;
            bf.x = W2[kk * 64 + c2];
            bf.y = W2[(kk + 1) * 64 + c2];
            acc2[t] = __builtin_amdgcn_wmma_f32_16x16x4_f32(
                false, a, false, bf, (short)0, acc2[t], false, false);
        }
    }

    // ---- bias + GELU + max over the 16 neighbors (rows of C2) -------------
#pragma unroll
    for (int t = 0; t < 4; ++t) {
        const float bb = b2[t * 16 + col];
        float m = -3.0e38f;
#pragma unroll
        for (int r = 0; r < 8; ++r) {
            const float x = acc2[t][r] + bb;    // rows M=r (+8 for upper half)
            m = fmaxf(m, gelu_exact(x));
        }
        // fold rows 0-7 (lanes 0-15) with rows 8-15 (lanes 16-31)
        m = fmaxf(m, __shfl_xor(m, 16));
        if (lane < 16) out[(size_t)p * 64 + t * 16 + lane] = m;
    }
}

// ------------------------------- launch -------------------------------------
extern "C" void kernel_launch(void* const* d_in, const int* in_sizes, int n_in,
                              void* d_out, int out_size, void* d_ws, size_t ws_size,
                              hipStream_t stream)
{
    const float* feats  = (const float*)d_in[0];  // (B, N, 64)
    const float* coords = (const float*)d_in[1];  // (B, N, 3)
    const float* W1     = (const float*)d_in[2];  // (131, 64)
    const float* b1     = (const float*)d_in[3];  // (64,)
    const float* W2     = (const float*)d_in[4];  // (64, 64)
    const float* b2     = (const float*)d_in[5];  // (64,)
    float*       out    = (float*)d_out;          // (B, N, 64)

    // Workspace layout: [ idx : B*N*16 ints = 2 MB ][ W1p : 132*64 floats ]
    int*   idx = (int*)d_ws;
    float* W1p = (float*)((char*)d_ws + (size_t)NBATCH * NPTS * KNN * sizeof(int));

    dim3 g1(NPTS / 256, NBATCH);
    knn_kernel<<<g1, 256, 0, stream>>>(coords, idx);

    w1pad_kernel<<<(EDGE_K * 64 + 255) / 256, 256, 0, stream>>>(W1, W1p);

    const int total_points = NBATCH * NPTS;       // 32768, one wave per point
    edgemlp_kernel<<<total_points / WAVES, 32 * WAVES, 0, stream>>>(
        feats, coords, idx, W1p, b1, W2, b2, out);
}